// GNN_27187142983846
// MI455X (gfx1250) — compile-verified
//
#include <hip/hip_runtime.h>
#include <hip/hip_bf16.h>

// ---------------------------------------------------------------------------
// GNN forward (3x NNConv-ish layers + BN/ReLU + mean-pool + FC) for gfx1250.
// f32 everywhere (matches reference). GEMMs use V_WMMA_F32_16X16X4_F32.
// ---------------------------------------------------------------------------

#define N_NODES    50000
#define N_EDGES    640000
#define HID        128
#define EDGE_DIM   32
#define OUT_DIM    2
#define NUM_GRAPHS 512
#define BN_EPS     1e-5f

typedef __attribute__((ext_vector_type(2))) float v2f;
typedef __attribute__((ext_vector_type(8))) float v8f;

// D = A(16x4,f32) * B(4x16,f32) + C(16x16,f32)
__device__ __forceinline__ v8f wmma_k4(v2f a, v2f b, v8f c) {
    return __builtin_amdgcn_wmma_f32_16x16x4_f32(
        /*neg_a=*/false, a, /*neg_b=*/false, b,
        /*c_mod=*/(short)0, c, /*reuse_a=*/false, /*reuse_b=*/false);
}

// ---------------------------------------------------------------------------
// H = X @ W + bias ; AGG = H  (self-loop contribution pre-seeded)
// grid.x = M/16 tiles, block = 256 (8 waves, wave w -> N-tile w of 128 cols)
// ---------------------------------------------------------------------------
__global__ void node_gemm_kernel(const float* __restrict__ X,
                                 const float* __restrict__ W,
                                 const float* __restrict__ bias,
                                 float* __restrict__ H,
                                 float* __restrict__ AGG) {
    const int mtile = blockIdx.x;
    const int wave  = threadIdx.x >> 5;      // 0..7  -> N tile
    const int lane  = threadIdx.x & 31;
    const int half  = lane >> 4;             // 0: K 0,1 | 1: K 2,3
    const int r     = lane & 15;

    const float* __restrict__ Arow = X + (size_t)(mtile * 16 + r) * HID;
    const int col = wave * 16 + r;           // this lane's B / output column

    v8f acc = {};
    #pragma unroll 4
    for (int k = 0; k < HID; k += 4) {
        v2f a, b;
        a.x = Arow[k + 2 * half];
        a.y = Arow[k + 2 * half + 1];
        b.x = W[(size_t)(k + 2 * half) * HID + col];
        b.y = W[(size_t)(k + 2 * half + 1) * HID + col];
        acc = wmma_k4(a, b, acc);
    }

    const float bv = bias[col];
    #pragma unroll
    for (int v = 0; v < 8; ++v) {
        const int row = mtile * 16 + v + 8 * half;   // C/D layout: VGPR v -> M=v(+8)
        const float o = acc[v] + bv;
        H  [(size_t)row * HID + col] = o;
        AGG[(size_t)row * HID + col] = o;
    }
}

// ---------------------------------------------------------------------------
// For each edge tile (16 edges): e = edge_attr @ We  (WMMA, K=32),
// msg = e + be + H[src], AGG[dst] += msg (f32 atomics, L2-resident).
// grid.x = E/16, block = 256 (8 waves over the 128 features)
// ---------------------------------------------------------------------------
__global__ void edge_msg_kernel(const float* __restrict__ EA,
                                const float* __restrict__ We,
                                const float* __restrict__ be,
                                const float* __restrict__ H,
                                const int*  __restrict__ src,
                                const int*  __restrict__ dst,
                                float* __restrict__ AGG) {
    const int etile = blockIdx.x;
    const int wave  = threadIdx.x >> 5;
    const int lane  = threadIdx.x & 31;
    const int half  = lane >> 4;
    const int r     = lane & 15;

    const float* __restrict__ Arow = EA + (size_t)(etile * 16 + r) * EDGE_DIM;
    const int col = wave * 16 + r;

    v8f acc = {};
    #pragma unroll
    for (int k = 0; k < EDGE_DIM; k += 4) {
        v2f a, b;
        a.x = Arow[k + 2 * half];
        a.y = Arow[k + 2 * half + 1];
        b.x = We[(size_t)(k + 2 * half) * HID + col];
        b.y = We[(size_t)(k + 2 * half + 1) * HID + col];
        acc = wmma_k4(a, b, acc);
    }

    const float bv = be[col];
    #pragma unroll
    for (int v = 0; v < 8; ++v) {
        const int e = etile * 16 + v + 8 * half;
        const int s = src[e];
        const int d = dst[e];
        const float msg = acc[v] + bv + H[(size_t)s * HID + col];
        atomicAdd(&AGG[(size_t)d * HID + col], msg);
    }
}

// ---------------------------------------------------------------------------
// Batchnorm pass 1: per-feature partial sum / sumsq (coalesced), atomic merge.
// grid 200 x block 128 ; each block covers 250 rows.
// ---------------------------------------------------------------------------
__global__ void bn_partial_kernel(const float* __restrict__ X,
                                  float* __restrict__ sumb,
                                  float* __restrict__ sqb) {
    const int f  = threadIdx.x;            // feature
    const int r0 = blockIdx.x * 250;
    float s = 0.f, s2 = 0.f;
    for (int rr = 0; rr < 250; ++rr) {
        const float v = X[(size_t)(r0 + rr) * HID + f];
        s += v; s2 += v * v;
    }
    atomicAdd(&sumb[f], s);
    atomicAdd(&sqb[f], s2);
}

// Batchnorm pass 2 + ReLU, elementwise. grid = N*HID/256.
__global__ void bn_apply_relu_kernel(const float* __restrict__ X,
                                     const float* __restrict__ sumb,
                                     const float* __restrict__ sqb,
                                     const float* __restrict__ gamma,
                                     const float* __restrict__ beta,
                                     float* __restrict__ Y) {
    const int idx = blockIdx.x * blockDim.x + threadIdx.x;
    const int f = idx & (HID - 1);
    const float inv_n = 1.0f / (float)N_NODES;
    const float mean = sumb[f] * inv_n;
    const float var  = sqb[f] * inv_n - mean * mean;
    const float y = (X[idx] - mean) * (gamma[f] * rsqrtf(var + BN_EPS)) + beta[f];
    Y[idx] = fmaxf(y, 0.0f);
}

// Mean-pool accumulate: one block per node, thread = feature.
__global__ void pool_acc_kernel(const float* __restrict__ Hf,
                                const int* __restrict__ batch,
                                float* __restrict__ psums,
                                float* __restrict__ cnt) {
    const int node = blockIdx.x;
    const int f = threadIdx.x;
    const int g = batch[node];
    atomicAdd(&psums[(size_t)g * HID + f], Hf[(size_t)node * HID + f]);
    if (f == 0) atomicAdd(&cnt[g], 1.0f);
}

// out[g] = (psums[g]/max(cnt,1)) @ Wfc + bfc ; block = 128, grid = NUM_GRAPHS.
__global__ void final_fc_kernel(const float* __restrict__ psums,
                                const float* __restrict__ cnt,
                                const float* __restrict__ Wfc,
                                const float* __restrict__ bfc,
                                float* __restrict__ out) {
    __shared__ float red0[HID];
    __shared__ float red1[HID];
    const int g = blockIdx.x;
    const int t = threadIdx.x;
    const float c = fmaxf(cnt[g], 1.0f);
    const float p = psums[(size_t)g * HID + t] / c;
    red0[t] = p * Wfc[t * OUT_DIM + 0];
    red1[t] = p * Wfc[t * OUT_DIM + 1];
    __syncthreads();
    for (int s = HID / 2; s > 0; s >>= 1) {
        if (t < s) { red0[t] += red0[t + s]; red1[t] += red1[t + s]; }
        __syncthreads();
    }
    if (t == 0) {
        out[g * OUT_DIM + 0] = red0[0] + bfc[0];
        out[g * OUT_DIM + 1] = red1[0] + bfc[1];
    }
}

// ---------------------------------------------------------------------------
extern "C" void kernel_launch(void* const* d_in, const int* in_sizes, int n_in,
                              void* d_out, int out_size, void* d_ws, size_t ws_size,
                              hipStream_t stream) {
    (void)in_sizes; (void)n_in; (void)out_size; (void)ws_size;

    const float* x     = (const float*)d_in[0];
    const float* ea    = (const float*)d_in[1];
    const int*   ei    = (const int*)  d_in[2];
    const int*   batch = (const int*)  d_in[3];
    const float* Wn[3] = { (const float*)d_in[4],  (const float*)d_in[8],  (const float*)d_in[12] };
    const float* bn[3] = { (const float*)d_in[5],  (const float*)d_in[9],  (const float*)d_in[13] };
    const float* We[3] = { (const float*)d_in[6],  (const float*)d_in[10], (const float*)d_in[14] };
    const float* be[3] = { (const float*)d_in[7],  (const float*)d_in[11], (const float*)d_in[15] };
    const float* gm[3] = { (const float*)d_in[16], (const float*)d_in[18], (const float*)d_in[20] };
    const float* bt[3] = { (const float*)d_in[17], (const float*)d_in[19], (const float*)d_in[21] };
    const float* Wfc   = (const float*)d_in[22];
    const float* bfc   = (const float*)d_in[23];
    float* out = (float*)d_out;

    const int* src = ei;               // edge_index[0]
    const int* dst = ei + N_EDGES;     // edge_index[1]

    const size_t NB = (size_t)N_NODES * HID;   // 6,400,000 floats
    float* ws    = (float*)d_ws;
    float* Xbuf  = ws;                 // BN output / next-layer input
    float* Hbuf  = ws + NB;            // node_fc output
    float* AGG   = ws + 2 * NB;        // aggregated messages (+ self loop)
    float* sumb  = ws + 3 * NB;        // [HID]
    float* sqb   = sumb + HID;         // [HID]
    float* psums = sqb + HID;          // [NUM_GRAPHS*HID]
    float* cnt   = psums + (size_t)NUM_GRAPHS * HID;  // [NUM_GRAPHS]

    const dim3 blk256(256), blk128(128);
    const int mtiles = N_NODES / 16;   // 3125
    const int etiles = N_EDGES / 16;   // 40000

    const float* layer_in = x;
    for (int L = 0; L < 3; ++L) {
        node_gemm_kernel<<<mtiles, blk256, 0, stream>>>(layer_in, Wn[L], bn[L], Hbuf, AGG);
        edge_msg_kernel<<<etiles, blk256, 0, stream>>>(ea, We[L], be[L], Hbuf, src, dst, AGG);
        hipMemsetAsync(sumb, 0, 2 * HID * sizeof(float), stream);
        bn_partial_kernel<<<200, blk128, 0, stream>>>(AGG, sumb, sqb);
        bn_apply_relu_kernel<<<(int)(NB / 256), blk256, 0, stream>>>(
            AGG, sumb, sqb, gm[L], bt[L], Xbuf);
        layer_in = Xbuf;
    }

    hipMemsetAsync(psums, 0, ((size_t)NUM_GRAPHS * HID + NUM_GRAPHS) * sizeof(float), stream);
    pool_acc_kernel<<<N_NODES, blk128, 0, stream>>>(Xbuf, batch, psums, cnt);
    final_fc_kernel<<<NUM_GRAPHS, blk128, 0, stream>>>(psums, cnt, Wfc, bfc, out);
}